// YoloV1Loss_21895743275708
// MI455X (gfx1250) — compile-verified
//
#include <hip/hip_runtime.h>
#include <cstdint>

// YOLOv1 loss, MI455X / gfx1250.
//   S=7 (49 cells), B=2, C=20, n_gt=16.  outputs: (batch,7,7,30) f32,
//   gt_boxes: (batch,16,4) f32, gt_labels: (batch,) int.  out: 1 f32 scalar.
//
// Design: 1 thread per (image, cell); 5 images per 256-thread block (245
// active lanes). GT boxes staged into LDS via CDNA5 async-load-to-LDS
// (ASYNCcnt path), predictions streamed with non-temporal b64 loads (read
// exactly once; don't pollute L2), block reduction via exact fp32
// V_WMMA_F32_16X16X4_F32 (D = A*ones + 0), deterministic 2-stage global
// reduction (no float atomics).

#define YCELLS 49
#define YB     2
#define YC     20
#define YNGT   16
#define ROWF   30      // floats per cell row (B*5 + C)
#define IPB    5       // batch items per block (5*49 = 245 active lanes)
#define EPSF   1e-9f

typedef float v2f __attribute__((ext_vector_type(2)));
typedef float v8f __attribute__((ext_vector_type(8)));

#if defined(__gfx1250__) && __has_builtin(__builtin_amdgcn_wmma_f32_16x16x4_f32)
#define USE_WMMA_REDUCE 1
#endif

// Exact fp32 wave32 sum using the matrix pipe: A is 16x4 f32 (2 VGPRs:
// VGPR0 = lane value at K=0 (lanes 0-15) / K=2 (lanes 16-31), VGPR1 = 0),
// B = ones(4x16).  D[m][n] = val[m] + val[m+16] broadcast over n.  Summing
// the 8 D VGPRs leaves identical half-wave partials in lanes 0-15 / 16-31;
// a second identical WMMA folds the two halves -> total in every lane.
// EXEC must be all-1s at the call site (callers are fully predicated).
__device__ __forceinline__ float wave_reduce_sum(float val) {
#ifdef USE_WMMA_REDUCE
  v2f a;   a.x = val;  a.y = 0.0f;
  v2f one; one.x = 1.0f; one.y = 1.0f;
  v8f cz = {};
  v8f d = __builtin_amdgcn_wmma_f32_16x16x4_f32(false, a, false, one,
                                                (short)0, cz, false, false);
  float s = d[0] + d[1] + d[2] + d[3] + d[4] + d[5] + d[6] + d[7];
  v2f a2; a2.x = s; a2.y = 0.0f;
  v8f d2 = __builtin_amdgcn_wmma_f32_16x16x4_f32(false, a2, false, one,
                                                 (short)0, cz, false, false);
  return d2[0];
#else
  #pragma unroll
  for (int off = 16; off > 0; off >>= 1) val += __shfl_xor(val, off, 32);
  return val;
#endif
}

// Block (256 thread = 8 wave) sum; result valid on thread 0.
__device__ __forceinline__ float block_reduce_sum_256(float val) {
  __shared__ float sred[8];
  float w = wave_reduce_sum(val);
  const int lane = threadIdx.x & 31;
  const int wid  = threadIdx.x >> 5;
  if (lane == 0) sred[wid] = w;
  __syncthreads();
  float tot = 0.0f;
  if (threadIdx.x == 0) {
    #pragma unroll
    for (int i = 0; i < 8; ++i) tot += sred[i];
  }
  return tot;
}

__global__ __launch_bounds__(256) void yolo_loss_main(
    const float* __restrict__ outs,      // (batch, 49, 30)
    const float* __restrict__ gtb,       // (batch, 16, 4)
    const int*   __restrict__ gtl,       // (batch,)
    float*       __restrict__ partials,  // (gridDim.x,)
    int batch)
{
  __shared__ __align__(16) float sgt[IPB * YNGT * 4];   // 320 floats, 1280 B
  const int t = threadIdx.x;

  // ---- Stage GT boxes for this block's 5 images into LDS via the CDNA5
  // ---- async-copy path (ASYNCcnt).  80 lanes x 16 B = 1280 B.
  {
    const long long gbase = (long long)blockIdx.x * (IPB * YNGT * 4);
    const long long gtot  = (long long)batch * (YNGT * 4);
    if (t < (IPB * YNGT * 4) / 4) {
      const long long off = gbase + (long long)t * 4;
      if (off + 4 <= gtot) {
        const float* src = gtb + off;
        // Flat address of LDS: low 32 bits are the workgroup-relative offset.
        unsigned ldsoff = (unsigned)(uintptr_t)(&sgt[t * 4]);
        asm volatile("global_load_async_to_lds_b128 %0, %1, off"
                     :: "v"(ldsoff), "v"(src) : "memory");
      }
    }
    asm volatile("s_wait_asynccnt 0" ::: "memory");
  }
  __syncthreads();

  // ---- Per-thread cell assignment (fully predicated, no early exit:
  // ---- WMMA below requires EXEC all-1s).
  const int  bl     = t / YCELLS;                 // local image 0..4 (5+ = idle)
  const int  cell   = t - bl * YCELLS;            // 0..48
  const bool active = (t < IPB * YCELLS);
  const int  bls    = active ? bl : 0;            // clamp for LDS indexing
  const long long b = (long long)blockIdx.x * IPB + bls;
  const bool valid  = active && (b < batch);
  const long long bc = valid ? b : 0;             // clamp for global loads

  // ---- Stream this cell's 30 floats (row is 8-B aligned -> b64 loads).
  // ---- Read-once data: non-temporal hint keeps the 96 MB stream out of L2.
  const float* po = outs + (bc * YCELLS + cell) * ROWF;
  float v[ROWF];
  #pragma unroll
  for (int i = 0; i < ROWF / 2; ++i) {
    v2f t2 = __builtin_nontemporal_load(
        reinterpret_cast<const v2f*>(po + 2 * i));
    v[2 * i] = t2.x;  v[2 * i + 1] = t2.y;
  }

  // Pred boxes (cx, cy, w, h, conf) x 2.
  const float b0x = v[0], b0y = v[1], b0w = v[2], b0h = v[3], b0c = v[4];
  const float b1x = v[5], b1y = v[6], b1w = v[7], b1h = v[8], b1c = v[9];
  const float a0 = b0w * b0h, a1 = b1w * b1h;
  const float x0l = b0x - 0.5f * b0w, x0r = b0x + 0.5f * b0w;
  const float y0l = b0y - 0.5f * b0h, y0r = b0y + 0.5f * b0h;
  const float x1l = b1x - 0.5f * b1w, x1r = b1x + 0.5f * b1w;
  const float y1l = b1y - 0.5f * b1h, y1r = b1y + 0.5f * b1h;

  // ---- Max/argmax IoU over 16 GT boxes (strict > replicates first-index
  // ---- argmax); keep both boxes' IoU at the winning GT for pbstar.
  float best = -1.0f, bi0 = 0.0f, bi1 = 0.0f;
  const float* g = &sgt[bls * (YNGT * 4)];
  #pragma unroll
  for (int j = 0; j < YNGT; ++j) {
    float4 gb = *reinterpret_cast<const float4*>(g + j * 4);   // ds_load_b128
    const float gxl = gb.x - 0.5f * gb.z, gxr = gb.x + 0.5f * gb.z;
    const float gyl = gb.y - 0.5f * gb.w, gyr = gb.y + 0.5f * gb.w;
    const float ga  = gb.z * gb.w;
    float iw0 = fmaxf(fminf(x0r, gxr) - fmaxf(x0l, gxl), 0.0f);
    float ih0 = fmaxf(fminf(y0r, gyr) - fmaxf(y0l, gyl), 0.0f);
    float in0 = iw0 * ih0;
    float io0 = in0 / (a0 + ga - in0 + EPSF);
    float iw1 = fmaxf(fminf(x1r, gxr) - fmaxf(x1l, gxl), 0.0f);
    float ih1 = fmaxf(fminf(y1r, gyr) - fmaxf(y1l, gyl), 0.0f);
    float in1 = iw1 * ih1;
    float io1 = in1 / (a1 + ga - in1 + EPSF);
    const float m = fmaxf(io0, io1);
    const bool u = m > best;
    bi0 = u ? io0 : bi0;  bi1 = u ? io1 : bi1;  best = u ? m : best;
  }
  const int pb = (bi0 >= bi1) ? 0 : 1;   // first-index argmax over B

  // Responsible pred box; NOTE: reference indexes gt_boxes with pbstar.
  const float px = pb ? b1x : b0x, py = pb ? b1y : b0y;
  const float pw = pb ? b1w : b0w, ph = pb ? b1h : b0h, pc = pb ? b1c : b0c;
  const float4 gs = *reinterpret_cast<const float4*>(g + pb * 4);

  const float dx = px - gs.x, dy = py - gs.y;
  const float dw = sqrtf(pw) - sqrtf(gs.z);
  const float dh = sqrtf(ph) - sqrtf(gs.w);
  const float loc   = 5.0f * (dx * dx + dy * dy + dw * dw + dh * dh);
  const float dcf   = pc - best;
  const float confl = dcf * dcf;

  // ---- Class NLL: mx + log(sum exp(c - mx)) - c[label]
  int lbl = __builtin_nontemporal_load(gtl + bc);
  lbl = (lbl < 0) ? 0 : ((lbl > YC - 1) ? (YC - 1) : lbl);
  float mx = v[10], xl = v[10];
  #pragma unroll
  for (int j = 1; j < YC; ++j) {
    const float cj = v[10 + j];
    mx = fmaxf(mx, cj);
    xl = (j == lbl) ? cj : xl;
  }
  float ss = 0.0f;
  #pragma unroll
  for (int j = 0; j < YC; ++j) ss += __expf(v[10 + j] - mx);
  const float cls = mx + __logf(ss) - xl;

  const float obj   = loc + confl + cls;
  const float noobj = 0.5f * (b0c * b0c + b1c * b1c);
  const float lv    = (best > 0.0f) ? obj : noobj;
  const float val   = valid ? lv : 0.0f;

  const float tot = block_reduce_sum_256(val);
  if (t == 0) partials[blockIdx.x] = tot;
}

// Deterministic second stage: fixed-order strided accumulation + WMMA tree.
__global__ __launch_bounds__(256) void yolo_loss_reduce(
    const float* __restrict__ p, float* __restrict__ out, int n, float scale)
{
  float acc = 0.0f;
  for (int i = threadIdx.x; i < n; i += 256) acc += p[i];
  const float tot = block_reduce_sum_256(acc);
  if (threadIdx.x == 0) out[0] = tot * scale;
}

extern "C" void kernel_launch(void* const* d_in, const int* in_sizes, int n_in,
                              void* d_out, int out_size, void* d_ws, size_t ws_size,
                              hipStream_t stream) {
  const float* outs = (const float*)d_in[0];   // (batch, 7, 7, 30) f32
  const float* gtb  = (const float*)d_in[1];   // (batch, 16, 4) f32
  const int*   gtl  = (const int*)d_in[2];     // (batch,) int

  const int batch = in_sizes[0] / (YCELLS * ROWF);   // 16384
  const int nb    = (batch + IPB - 1) / IPB;         // blocks (5 images each)
  float* partials = (float*)d_ws;                    // nb floats of scratch

  yolo_loss_main<<<nb, 256, 0, stream>>>(outs, gtb, gtl, partials, batch);
  yolo_loss_reduce<<<1, 256, 0, stream>>>(partials, (float*)d_out, nb,
                                          1.0f / (float)batch);
}